// BallClusterLearningLoss_4157528343141
// MI455X (gfx1250) — compile-verified
//
#include <hip/hip_runtime.h>
#include <math.h>

#define NROWS   262144
#define DDIM    128
#define KCLUST  256

#define P1_BLOCKS   256
#define P1_THREADS  512
#define P1_ROWS_PER_BLOCK (NROWS / P1_BLOCKS)                    // 1024
#define P1_ROWS_PER_WAVE  (P1_ROWS_PER_BLOCK / (P1_THREADS/32))  // 64
#define LS_TOTAL (KCLUST * DDIM + KCLUST)                        // 33024 floats (129 KB)

typedef float v2f __attribute__((ext_vector_type(2)));
typedef float v8f __attribute__((ext_vector_type(8)));
typedef int   v4i __attribute__((vector_size(16)));              // matches builtin proto
typedef __attribute__((address_space(1))) v4i* g_v4i_ptr;        // global
typedef __attribute__((address_space(3))) v4i* l_v4i_ptr;        // LDS

// ---- order-preserving float <-> uint map (for atomicMin on floats) ----
__device__ __forceinline__ unsigned fenc(float f) {
    unsigned u = __float_as_uint(f);
    return (u & 0x80000000u) ? ~u : (u | 0x80000000u);
}
__device__ __forceinline__ float fdec(unsigned k) {
    unsigned u = (k & 0x80000000u) ? (k & 0x7FFFFFFFu) : ~k;
    return __uint_as_float(u);
}

// ---- pass 0: zero centroid accumulators ----
__global__ void k_zero(float* __restrict__ p, int n) {
    int i = blockIdx.x * blockDim.x + threadIdx.x;
    if (i < n) p[i] = 0.0f;
}

// ---- pass 1: segment-sum centroids + per-row ||x||^2 ----
// CDNA5: whole K x D fp32 accumulator fits in the 320KB WGP LDS (129KB).
// One wave32 per row; ds_add_f32 into LDS; single atomic flush per block.
__global__ __launch_bounds__(P1_THREADS) void k_accum(const float* __restrict__ X,
                                                      const int* __restrict__ labels,
                                                      float* __restrict__ csum,   // [K*D] then [K] counts
                                                      float* __restrict__ x2) {
    __shared__ float ls[LS_TOTAL];   // [0,32768) sums, [32768,33024) counts
    int tid = threadIdx.x;
    for (int i = tid; i < LS_TOTAL; i += P1_THREADS) ls[i] = 0.0f;
    __syncthreads();

    int wv = tid >> 5, lane = tid & 31;
    int rowbase = blockIdx.x * P1_ROWS_PER_BLOCK + wv * P1_ROWS_PER_WAVE;

#pragma unroll 4
    for (int j = 0; j < P1_ROWS_PER_WAVE; ++j) {
        int row = rowbase + j;
        int lab = labels[row];
        float4 v = ((const float4*)(X + (size_t)row * DDIM))[lane];
        float ss = v.x * v.x + v.y * v.y + v.z * v.z + v.w * v.w;

        float* dst = &ls[lab * DDIM + lane * 4];
        atomicAdd(dst + 0, v.x);
        atomicAdd(dst + 1, v.y);
        atomicAdd(dst + 2, v.z);
        atomicAdd(dst + 3, v.w);

        for (int m = 16; m >= 1; m >>= 1) ss += __shfl_xor(ss, m, 32);
        if (lane == 0) {
            x2[row] = ss;
            atomicAdd(&ls[KCLUST * DDIM + lab], 1.0f);
        }
    }
    __syncthreads();

    // flush block-partial sums+counts (contiguous in ws) with global atomics
    for (int i = tid; i < LS_TOTAL; i += P1_THREADS)
        atomicAdd(&csum[i], ls[i]);
}

// ---- pass 2: mean + L2-normalize centroids, compute c2 (one block per cluster) ----
__global__ __launch_bounds__(128) void k_centroid(const float* __restrict__ csum,
                                                  const float* __restrict__ ccount,
                                                  float* __restrict__ cent,
                                                  float* __restrict__ c2) {
    __shared__ float red[4];
    int k = blockIdx.x;
    int t = threadIdx.x;           // 128 threads, one per dim
    int lane = t & 31, w = t >> 5;

    float cnt = ccount[k];
    float m = csum[(size_t)k * DDIM + t] / cnt;

    float ss = m * m;
    for (int o = 16; o >= 1; o >>= 1) ss += __shfl_xor(ss, o, 32);
    if (lane == 0) red[w] = ss;
    __syncthreads();
    float norm2 = red[0] + red[1] + red[2] + red[3];

    float v = m / sqrtf(norm2);
    cent[(size_t)k * DDIM + t] = v;

    float ss2 = v * v;
    for (int o = 16; o >= 1; o >>= 1) ss2 += __shfl_xor(ss2, o, 32);
    __syncthreads();
    if (lane == 0) red[w] = ss2;
    __syncthreads();
    if (t == 0) c2[k] = red[0] + red[1] + red[2] + red[3];
}

// ---- pass 3: fp32 WMMA distance tiles + per-row pos / min-neg losses ----
// block = 256 threads (8 waves), handles 16 rows x all 256 clusters.
__global__ __launch_bounds__(256) void k_dist(const float* __restrict__ X,
                                              const int* __restrict__ labels,
                                              const float* __restrict__ cent,
                                              const float* __restrict__ c2,
                                              const float* __restrict__ x2,
                                              const float* __restrict__ h_bias,
                                              float* __restrict__ bpos,
                                              float* __restrict__ bneg) {
    __shared__ float    xs[16 * 132];   // X tile, padded pitch 132 (bank-conflict-free A reads)
    __shared__ int      lab[16];
    __shared__ float    x2s[16];
    __shared__ unsigned mkey[16];
    __shared__ float    posv[16];
    __shared__ float    plA[16], nlA[16];

    int tid = threadIdx.x;
    int rowbase = blockIdx.x * 16;

    // stage 16x128 X tile into LDS (2 x 16B per thread)
#if defined(__gfx1250__) && __has_builtin(__builtin_amdgcn_global_load_async_to_lds_b128)
    for (int i = tid; i < 512; i += 256) {
        int r = i >> 5;
        int c = (i & 31) * 4;
        __builtin_amdgcn_global_load_async_to_lds_b128(
            (g_v4i_ptr)(X + (size_t)(rowbase + r) * DDIM + c),
            (l_v4i_ptr)&xs[r * 132 + c],
            /*offset=*/0, /*cpol=*/0);
    }
#else
    for (int i = tid; i < 512; i += 256) {
        int r = i >> 5;
        int c = (i & 31) * 4;
        float4 v = *(const float4*)(X + (size_t)(rowbase + r) * DDIM + c);
        *(float4*)(&xs[r * 132 + c]) = v;
    }
#endif
    if (tid < 16) {
        lab[tid]  = labels[rowbase + tid];
        x2s[tid]  = x2[rowbase + tid];
        mkey[tid] = 0xFFFFFFFFu;       // encodes "+inf"
        posv[tid] = 0.0f;
    }
#if defined(__gfx1250__) && __has_builtin(__builtin_amdgcn_global_load_async_to_lds_b128)
#if __has_builtin(__builtin_amdgcn_s_wait_asynccnt)
    __builtin_amdgcn_s_wait_asynccnt(0);
#else
    asm volatile("s_wait_asynccnt 0x0" ::: "memory");
#endif
#endif
    __syncthreads();

    int wave = tid >> 5, lane = tid & 31;
    int hi = lane >> 4;                // 0: K-pair {0,1}; 1: K-pair {2,3} in each 4-step
    int ln = lane & 15;                // A: row M; B: column N
    int col0 = wave * 16 + ln;         // this wave's two 16-column tiles
    int col1 = (wave + 8) * 16 + ln;

    v8f acc0 = {0.f, 0.f, 0.f, 0.f, 0.f, 0.f, 0.f, 0.f};
    v8f acc1 = {0.f, 0.f, 0.f, 0.f, 0.f, 0.f, 0.f, 0.f};
    const float* c0p = cent + (size_t)col0 * DDIM;
    const float* c1p = cent + (size_t)col1 * DDIM;

    for (int ks = 0; ks < 32; ++ks) {
        int kb = ks * 4 + hi * 2;
        v2f a  = *(const v2f*)(&xs[ln * 132 + kb]);   // A: X[M=ln][kb, kb+1]   (LDS b64)
        v2f b0 = *(const v2f*)(c0p + kb);             // B: cent[N=col0][kb, kb+1]
        v2f b1 = *(const v2f*)(c1p + kb);
        acc0 = __builtin_amdgcn_wmma_f32_16x16x4_f32(false, a, false, b0,
                                                     (short)0, acc0, false, false);
        acc1 = __builtin_amdgcn_wmma_f32_16x16x4_f32(false, a, false, b1,
                                                     (short)0, acc1, false, false);
    }

    // epilogue: distance = ||x||^2 + ||c||^2 - 2 x.c ; split pos vs min-neg
    float c2v0 = c2[col0];
    float c2v1 = c2[col1];
    for (int r = 0; r < 8; ++r) {
        int rl = r + hi * 8;           // local row index of C/D element in VGPR r
        float d0 = x2s[rl] + c2v0 - 2.0f * acc0[r];
        float d1 = x2s[rl] + c2v1 - 2.0f * acc1[r];
        int lb = lab[rl];
        if (lb == col0) posv[rl] = d0;                       // unique writer per row
        else            atomicMin(&mkey[rl], fenc(d0));
        if (lb == col1) posv[rl] = d1;
        else            atomicMin(&mkey[rl], fenc(d1));
    }
    __syncthreads();

    if (tid < 16) {
        float h    = h_bias[0];
        float bias = log1pf(expf(h));                        // softplus
        float posb = bias;
        float negb = 9.0f * bias + 0.05f;                    // GAMMA_EPS
        float mn   = fdec(mkey[tid]);
        plA[tid] = fmaxf(posv[tid] - posb, 0.0f);
        nlA[tid] = fmaxf(negb - mn, 0.0f);
    }
    __syncthreads();
    if (tid == 0) {
        float sp = 0.f, sn = 0.f;
        for (int i = 0; i < 16; ++i) { sp += plA[i]; sn += nlA[i]; }
        bpos[blockIdx.x] = sp;
        bneg[blockIdx.x] = sn;
    }
}

// ---- pass 4: deterministic final reduction + scaling ----
__global__ __launch_bounds__(256) void k_final(const float* __restrict__ bpos,
                                               const float* __restrict__ bneg,
                                               float* __restrict__ out, int nblocks) {
    __shared__ float sp[256], sn[256];
    int t = threadIdx.x;
    float ap = 0.f, an = 0.f;
    for (int i = t; i < nblocks; i += 256) { ap += bpos[i]; an += bneg[i]; }
    sp[t] = ap; sn[t] = an;
    __syncthreads();
    for (int s = 128; s > 0; s >>= 1) {
        if (t < s) { sp[t] += sp[t + s]; sn[t] += sn[t + s]; }
        __syncthreads();
    }
    if (t == 0) {
        out[0] = 4.0f * sp[0] / (float)NROWS;   // ALPHA_POS * mean(pos)
        out[1] = 1.0f * sn[0] / (float)NROWS;   // ALPHA_NEG * mean(neg)
    }
}

extern "C" void kernel_launch(void* const* d_in, const int* in_sizes, int n_in,
                              void* d_out, int out_size, void* d_ws, size_t ws_size,
                              hipStream_t stream) {
    const float* X      = (const float*)d_in[0];
    // d_in[1] = scores (unused by the loss)
    const int*   labels = (const int*)d_in[2];
    const float* h_bias = (const float*)d_in[3];
    // d_in[4] = K (hardcoded 256)

    float* ws     = (float*)d_ws;
    float* csum   = ws;                      // 32768 sums + 256 counts (contiguous)
    float* ccount = csum   + KCLUST * DDIM;  // 256
    float* cent   = ccount + KCLUST;         // 32768
    float* c2     = cent   + KCLUST * DDIM;  // 256
    float* x2     = c2     + KCLUST;         // 262144
    float* bpos   = x2     + NROWS;          // 16384
    float* bneg   = bpos   + NROWS / 16;     // 16384

    int zn = LS_TOTAL;                       // csum + ccount
    k_zero<<<(zn + 255) / 256, 256, 0, stream>>>(csum, zn);
    k_accum<<<P1_BLOCKS, P1_THREADS, 0, stream>>>(X, labels, csum, x2);
    k_centroid<<<KCLUST, 128, 0, stream>>>(csum, ccount, cent, c2);
    int nb3 = NROWS / 16;
    k_dist<<<nb3, 256, 0, stream>>>(X, labels, cent, c2, x2, h_bias, bpos, bneg);
    k_final<<<1, 256, 0, stream>>>(bpos, bneg, (float*)d_out, nb3);
}